// Model_39788577030287
// MI455X (gfx1250) — compile-verified
//
#include <hip/hip_runtime.h>
#include <hip/hip_bf16.h>
#include <math.h>

// ---------------------------------------------------------------------------
// Model: person-graph tag head.
//   N=64, C=4, T=300, V=17, M=10, Cf=256, Tf=75, num_class=60, K=4
// Dominant cost: streaming feat (835 MB) -> pf (~36 us at 23.3 TB/s).
// Everything else is L2-resident and tiny; proj GEMM runs on WMMA f32.
// ---------------------------------------------------------------------------

#define N_B   64
#define CF    256
#define TF    75
#define VJ    17
#define MP    10
#define MPAD  16              // pf padded to 16 person rows (rows 10..15 zero)
#define NCLS  60
#define TVM   (TF * VJ * MP)  // 12750 floats per (n, c)
#define HIP_L 11
#define HIP_R 12

typedef __attribute__((ext_vector_type(2))) float v2f;
typedef __attribute__((ext_vector_type(8))) float v8f;

// ---------------------------------------------------------------------------
// Kernel 1: pf[n][m][c] = mean_{t,v} feat[n][c][t][v][m]   (rows m>=10 zeroed)
// One block per (n, c): 12750 contiguous floats. 320 threads, float2 loads
// (8B-aligned: block offset 12750*c is even).  Stride 640 == 0 (mod 10), so
// each LDS slot maps to a fixed person index m = slot % 10.
// ---------------------------------------------------------------------------
__global__ __launch_bounds__(320)
void reduce_feat_kernel(const float* __restrict__ feat, float* __restrict__ pf) {
  const int bc = blockIdx.x;          // n*CF + c
  const int n  = bc >> 8;
  const int c  = bc & 255;
  const float* base = feat + (size_t)bc * TVM;
  const int tid = threadIdx.x;        // 0..319

  float acc0 = 0.f, acc1 = 0.f;
  for (int i = 2 * tid; i + 1 < TVM; i += 640) {
    float2 v = *(const float2*)(base + i);   // global_load_b64, coalesced
    acc0 += v.x;
    acc1 += v.y;
  }

  __shared__ float red[640];
  red[2 * tid]     = acc0;
  red[2 * tid + 1] = acc1;
  __syncthreads();

  if (tid < MPAD) {
    float s = 0.f;
    if (tid < MP) {
      #pragma unroll 8
      for (int j = tid; j < 640; j += MP) s += red[j];   // slot j -> m = j % 10
      s *= (1.0f / (float)(TF * VJ));
    }
    pf[((size_t)n * MPAD + tid) * CF + c] = s;           // zero pad rows 10..15
  }
}

// ---------------------------------------------------------------------------
// Kernel 2: pos (mid-hip, time-mean)  ->  row-stochastic kNN adjacency A[n][10][10]
// One block per batch element. Rank-count selection reproduces
// jax.lax.top_k(-d, 4) tie-breaking (ascending d, then ascending index).
// ---------------------------------------------------------------------------
__global__ __launch_bounds__(128)
void build_adj_kernel(const float* __restrict__ x, float* __restrict__ Aout) {
  const int n   = blockIdx.x;
  const int tid = threadIdx.x;

  __shared__ float pos[MP][3];
  __shared__ float dmat[MP][MP];
  __shared__ float amat[MP][MP];
  __shared__ float rowsum[MP];

  if (tid < 3 * MP) {
    const int m = tid % MP, c = tid / MP;
    // x index: (((n*4 + c)*300 + t)*17 + v)*10 + m
    const float* xl = x + ((((size_t)n * 4 + c) * 300) * VJ + HIP_L) * MP + m;
    const float* xr = x + ((((size_t)n * 4 + c) * 300) * VJ + HIP_R) * MP + m;
    float s = 0.f;
    for (int t = 0; t < 300; ++t)
      s += xl[(size_t)t * (VJ * MP)] + xr[(size_t)t * (VJ * MP)];
    pos[m][c] = s * (0.5f / 300.0f);
  }
  __syncthreads();

  if (tid < MP * MP) {
    const int i = tid / MP, j = tid % MP;
    const float dx = pos[i][0] - pos[j][0];
    const float dy = pos[i][1] - pos[j][1];
    const float dz = pos[i][2] - pos[j][2];
    dmat[i][j] = sqrtf(dx * dx + dy * dy + dz * dz);
  }
  __syncthreads();

  if (tid < MP * MP) {
    const int i = tid / MP, j = tid % MP;
    const float dij = dmat[i][j];
    int rank = 0;
    #pragma unroll
    for (int jp = 0; jp < MP; ++jp) {
      const float djp = dmat[i][jp];
      if (djp < dij || (djp == dij && jp < j)) rank++;
    }
    amat[i][j] = (rank < 4 ? 1.0f : 0.0f) + (i == j ? 1.0f : 0.0f);
  }
  __syncthreads();

  if (tid < MP) {
    float s = 0.f;
    #pragma unroll
    for (int j = 0; j < MP; ++j) s += amat[tid][j];
    rowsum[tid] = s + 1e-6f;
  }
  __syncthreads();

  if (tid < MP * MP)
    Aout[(size_t)n * (MP * MP) + tid] = amat[tid / MP][tid % MP] / rowsum[tid / MP];
}

// ---------------------------------------------------------------------------
// Kernel 3: WMMA proj GEMM + graph-conv fuse + ReLU + mean over persons.
//   h = pf @ proj_w^T + proj_b            (640 x 256 GEMM via v_wmma_f32_16x16x4_f32)
//   h = relu(h + 0.1 * A @ h)             (all 10 persons live in one 16x16 tile)
//   hbar[n][c] = mean_m h                 (plain stores, column tiles partition c)
// One wave (32 threads) per (n, column-tile).  grid = 64 * 16.
// pf is zero-padded to 16 rows, so all fragment loads are unconditional b64
// and control flow stays uniform (EXEC all-ones around every WMMA).
// ---------------------------------------------------------------------------
__global__ __launch_bounds__(32)
void proj_fuse_kernel(const float* __restrict__ pf, const float* __restrict__ proj_w,
                      const float* __restrict__ proj_b, const float* __restrict__ A,
                      float* __restrict__ hbar) {
  const int n    = blockIdx.x >> 4;
  const int c0   = (blockIdx.x & 15) << 4;
  const int lane = threadIdx.x;
  const int half = lane >> 4;       // K sub-block selector for A/B frags
  const int l16  = lane & 15;

  // A-matrix 16x4 f32 layout: lane<16 -> M=lane, K=0/1 ; lane>=16 -> M=lane-16, K=2/3
  const float* pfrow = pf + ((size_t)n * MPAD + l16) * CF;
  // B-matrix 4x16: B[k][col] = proj_w[col][k]
  const int col      = c0 + l16;
  const float* wrow  = proj_w + (size_t)col * CF;

  v8f acc = {0.f, 0.f, 0.f, 0.f, 0.f, 0.f, 0.f, 0.f};
  #pragma unroll 8
  for (int k = 0; k < CF; k += 4) {
    const int kk = k + 2 * half;                 // 8B-aligned (kk even)
    v2f a = *(const v2f*)(pfrow + kk);
    v2f b = *(const v2f*)(wrow + kk);
    acc = __builtin_amdgcn_wmma_f32_16x16x4_f32(
        /*neg_a=*/false, a, /*neg_b=*/false, b,
        /*c_mod=*/(short)0, acc, /*reuse_a=*/false, /*reuse_b=*/false);
  }

  __shared__ float hlds[16][16];     // h tile: [m][local col]
  __shared__ float Alds[MP * MP];
  __shared__ float partial_hi[16];

  const float bias = proj_b[col];
  // D layout: VGPR r holds M = r (lanes 0-15) / M = 8+r (lanes 16-31), N = l16
  #pragma unroll
  for (int r = 0; r < 8; ++r)
    hlds[r + 8 * half][l16] = acc[r] + bias;

  for (int idx = lane; idx < MP * MP; idx += 32)
    Alds[idx] = A[(size_t)n * (MP * MP) + idx];
  __syncthreads();

  float hp[MP];
  #pragma unroll
  for (int p = 0; p < MP; ++p) hp[p] = hlds[p][l16];

  float part = 0.f;
  #pragma unroll
  for (int r = 0; r < 8; ++r) {
    const int m = r + 8 * half;
    if (m < MP) {
      float s = 0.f;
      #pragma unroll
      for (int p = 0; p < MP; ++p) s += Alds[m * MP + p] * hp[p];
      part += fmaxf(hp[m] + 0.1f * s, 0.f);      // LAMBDA_FUSE = 0.1, ReLU
    }
  }

  if (half) partial_hi[l16] = part;              // m = 8..9 contribution
  __syncthreads();
  if (!half)
    hbar[(size_t)n * CF + col] = (part + partial_hi[l16]) * (1.0f / (float)MP);
}

// ---------------------------------------------------------------------------
// Kernel 4: out[n][j] = logits_base[n][j] + sigmoid(tag_scale) * (hbar[n]·cls_w[j] + cls_b[j])
// ---------------------------------------------------------------------------
__global__ __launch_bounds__(64)
void head_kernel(const float* __restrict__ hbar, const float* __restrict__ cls_w,
                 const float* __restrict__ cls_b, const float* __restrict__ logits_base,
                 const float* __restrict__ tag_scale, float* __restrict__ out) {
  const int n = blockIdx.x;
  const int j = threadIdx.x;
  if (j >= NCLS) return;
  const float* h = hbar  + (size_t)n * CF;
  const float* w = cls_w + (size_t)j * CF;
  float s = 0.f;
  #pragma unroll 8
  for (int c = 0; c < CF; ++c) s += h[c] * w[c];
  const float gate = 1.0f / (1.0f + expf(-tag_scale[0]));   // TAG_SCALE_MAX=1, RAMP=1
  out[(size_t)n * NCLS + j] = logits_base[(size_t)n * NCLS + j] + gate * (s + cls_b[j]);
}

// ---------------------------------------------------------------------------
extern "C" void kernel_launch(void* const* d_in, const int* in_sizes, int n_in,
                              void* d_out, int out_size, void* d_ws, size_t ws_size,
                              hipStream_t stream) {
  const float* x           = (const float*)d_in[0];  // (64,4,300,17,10)
  const float* feat        = (const float*)d_in[1];  // (64,256,75,17,10)
  const float* logits_base = (const float*)d_in[2];  // (64,60)
  const float* proj_w      = (const float*)d_in[3];  // (256,256)
  const float* proj_b      = (const float*)d_in[4];  // (256,)
  const float* cls_w       = (const float*)d_in[5];  // (60,256)
  const float* cls_b       = (const float*)d_in[6];  // (60,)
  const float* tag_scale   = (const float*)d_in[7];  // scalar
  float* out = (float*)d_out;

  float* ws   = (float*)d_ws;
  float* pf   = ws;                                  // N*MPAD*Cf = 262144 floats
  float* A    = ws + (size_t)N_B * MPAD * CF;        // N*M*M     = 6400 floats
  float* hbar = A  + (size_t)N_B * MP * MP;          // N*Cf      = 16384 floats

  reduce_feat_kernel<<<N_B * CF, 320, 0, stream>>>(feat, pf);
  build_adj_kernel  <<<N_B,      128, 0, stream>>>(x, A);
  proj_fuse_kernel  <<<N_B * 16,  32, 0, stream>>>(pf, proj_w, proj_b, A, hbar);
  head_kernel       <<<N_B,       64, 0, stream>>>(hbar, cls_w, cls_b, logits_base,
                                                   tag_scale, out);
}